// FullyConnectedGeometricProductBase_40965398069487
// MI455X (gfx1250) — compile-verified
//
#include <hip/hip_runtime.h>

typedef __attribute__((ext_vector_type(2))) float v2f;
typedef __attribute__((ext_vector_type(8))) float v8f;

#define BATCH    8192
#define NIN      64
#define NOUT     64
#define DCOMP    8
#define TB       16      // batch rows per block
#define NTHREADS 256     // 8 waves (wave32)

// LDS layout (float offsets)
#define XS_OFF   0       // xs[n][r][i]      : 64*16*8  = 8192 floats
#define XR_OFF   8192    // xr[o][r][i]      : 8192 floats; reused as out staging [r][m][j]
#define U_OFF    16384   // u[n][r][j*2+g]   : 64*16*16 = 16384 floats
#define LDS_FLOATS 32768 // 128 KB

__device__ __forceinline__ float sigmoidf_(float a) { return 1.0f / (1.0f + __expf(-a)); }

__global__ __launch_bounds__(NTHREADS)
void fcgp_kernel(const float* __restrict__ x,    // (B, 64, 8)
                 const float* __restrict__ wgt,  // (64, 64, 4)
                 const float* __restrict__ wr,   // (64, 64, 2)
                 const float* __restrict__ wl,   // (64, 64, 2)
                 const float* __restrict__ bl,   // (64)
                 const float* __restrict__ an,   // (64, 2)
                 float* __restrict__ out)        // (B, 64, 8)
{
  extern __shared__ __align__(16) float lds[];
  const int tid  = threadIdx.x;
  const int lane = tid & 31;
  const int wave = tid >> 5;      // 0..7  -> component / output blade
  const int col  = lane & 15;     // A: M row ; B,C/D: N column
  const int ksel = lane >> 4;     // selects K pair (A/B) or M half (C/D)

  // Keep the tiny weight set warm in cache (CDNA5 global_prefetch path).
  __builtin_prefetch(wgt, 0, 0);
  __builtin_prefetch(wr,  0, 0);
  __builtin_prefetch(wl,  0, 0);

  // ---------- Phase A: stage x tile -> LDS xs[n][r][i] ----------
  {
    const float4* xg = (const float4*)(x + (size_t)blockIdx.x * (TB * NIN * DCOMP));
    #pragma unroll
    for (int it = 0; it < 8; ++it) {
      const int t = tid + it * NTHREADS;     // 0..2047 float4s
      float4 v = xg[t];
      const int f  = t << 2;                 // flat float index (r*512 + n*8 + i)
      const int r  = f >> 9;
      const int n  = (f >> 3) & 63;
      const int i0 = f & 7;                  // 0 or 4
      *(float4*)&lds[XS_OFF + n * 128 + r * 8 + i0] = v;
    }
  }
  __syncthreads();

  // ---------- Phase B: xr_pre[o][r][i] = sum_n xs[n][r][i] * wr[o][n][g_i] ----------
  {
    const int comp = wave;
    const int g    = __popc(comp) & 1;
    const float* xsL = &lds[XS_OFF + col * 8 + comp];    // + n*128
    for (int ot = 0; ot < 4; ++ot) {
      v8f acc = {};
      const float* wrp = wr + (size_t)(ot * 16 + col) * (NIN * 2) + g;  // + k*2
      #pragma unroll
      for (int kc = 0; kc < 16; ++kc) {
        const int k0 = kc * 4 + ksel * 2;
        v2f a, b;
        a.x = xsL[(k0    ) * 128];
        a.y = xsL[(k0 + 1) * 128];
        b.x = wrp[(k0    ) * 2];
        b.y = wrp[(k0 + 1) * 2];
        acc = __builtin_amdgcn_wmma_f32_16x16x4_f32(false, a, false, b, (short)0, acc, false, false);
      }
      // C layout: element (M = v + 8*ksel, N = col) -> xr[o=ot*16+col][r][comp]
      float* xrL = &lds[XR_OFF + (ot * 16 + col) * 128 + ksel * 64 + comp];
      #pragma unroll
      for (int v = 0; v < 8; ++v) xrL[v * 8] = acc[v];
    }
  }
  __syncthreads();

  // ---------- Phase C: grade-group normalization (in place on xr) ----------
  {
    #pragma unroll
    for (int it = 0; it < 4; ++it) {
      const int t = tid + it * NTHREADS;   // (r,o) pair, 1024 total
      const int r = t & 15;
      const int o = t >> 4;
      float* p = &lds[XR_OFF + o * 128 + r * 8];
      float v0 = p[0], v1 = p[1], v2 = p[2], v3 = p[3];
      float v4 = p[4], v5 = p[5], v6 = p[6], v7 = p[7];
      // GRP = [0,1,1,0,1,0,0,1]
      const float q0 = v0*v0 + v3*v3 + v5*v5 + v6*v6;
      const float q1 = v1*v1 + v2*v2 + v4*v4 + v7*v7;
      const float s0 = sigmoidf_(an[o * 2 + 0]) * (sqrtf(q0) - 1.0f) + 1.0f;
      const float s1 = sigmoidf_(an[o * 2 + 1]) * (sqrtf(q1) - 1.0f) + 1.0f;
      const float r0 = 1.0f / s0, r1 = 1.0f / s1;
      p[0] = v0 * r0; p[3] = v3 * r0; p[5] = v5 * r0; p[6] = v6 * r0;
      p[1] = v1 * r1; p[2] = v2 * r1; p[4] = v4 * r1; p[7] = v7 * r1;
    }
  }
  __syncthreads();

  // ---------- Phase D: u[n][r][j,g] = sum_{i:GRP[i]=g} sign(i,j)*x[i]*xr[i^j] ----------
  {
    #pragma unroll
    for (int it = 0; it < 4; ++it) {
      const int t = tid + it * NTHREADS;   // (r,n) pair
      const int r = t & 15;
      const int n = t >> 4;
      float xv[8], rv[8];
      #pragma unroll
      for (int i = 0; i < 8; ++i) {
        xv[i] = lds[XS_OFF + n * 128 + r * 8 + i];
        rv[i] = lds[XR_OFF + n * 128 + r * 8 + i];
      }
      float* up = &lds[U_OFF + n * 256 + r * 16];
      #pragma unroll
      for (int j = 0; j < 8; ++j) {
        float u0 = 0.0f, u1 = 0.0f;
        #pragma unroll
        for (int i = 0; i < 8; ++i) {
          const int  sw  = __popc((i >> 1) & j) + __popc((i >> 2) & j); // Cayley swap count
          const float sg = (sw & 1) ? -1.0f : 1.0f;
          const float tm = sg * xv[i] * rv[i ^ j];
          if (__popc(i) & 1) u1 += tm; else u0 += tm;   // compile-time after unroll
        }
        up[j * 2 + 0] = u0;
        up[j * 2 + 1] = u1;
      }
    }
  }
  __syncthreads();

  // ---------- Phase E: out = (left_linear + bias + gp) / sqrt(2), fused in one acc ----------
  {
    const int j  = wave;
    const int gj = __popc(j) & 1;
    const float* xsL = &lds[XS_OFF + col * 8 + j];       // + n*128
    const float* uL  = &lds[U_OFF + col * 16 + j * 2];   // + n*256 (+1 for g=1)
    const float inv_sqrt2 = 0.70710678118654752440f;
    for (int mt = 0; mt < 4; ++mt) {
      const int m0 = mt * 16 + col;
      v8f acc = {};
      const float* wlp = wl  + (size_t)m0 * (NIN * 2) + gj;
      const float* wg0 = wgt + (size_t)m0 * (NIN * 4) + gj;       // path(0,gj) = gj
      const float* wg1 = wgt + (size_t)m0 * (NIN * 4) + 2 + gj;   // path(1,gj) = 2+gj
      #pragma unroll
      for (int kc = 0; kc < 16; ++kc) {                 // left linear
        const int k0 = kc * 4 + ksel * 2;
        v2f a, b;
        a.x = xsL[(k0    ) * 128];  a.y = xsL[(k0 + 1) * 128];
        b.x = wlp[(k0    ) * 2];    b.y = wlp[(k0 + 1) * 2];
        acc = __builtin_amdgcn_wmma_f32_16x16x4_f32(false, a, false, b, (short)0, acc, false, false);
      }
      #pragma unroll
      for (int kc = 0; kc < 16; ++kc) {                 // gp, g = 0
        const int k0 = kc * 4 + ksel * 2;
        v2f a, b;
        a.x = uL[(k0    ) * 256];   a.y = uL[(k0 + 1) * 256];
        b.x = wg0[(k0    ) * 4];    b.y = wg0[(k0 + 1) * 4];
        acc = __builtin_amdgcn_wmma_f32_16x16x4_f32(false, a, false, b, (short)0, acc, false, false);
      }
      #pragma unroll
      for (int kc = 0; kc < 16; ++kc) {                 // gp, g = 1
        const int k0 = kc * 4 + ksel * 2;
        v2f a, b;
        a.x = uL[(k0    ) * 256 + 1]; a.y = uL[(k0 + 1) * 256 + 1];
        b.x = wg1[(k0    ) * 4];      b.y = wg1[(k0 + 1) * 4];
        acc = __builtin_amdgcn_wmma_f32_16x16x4_f32(false, a, false, b, (short)0, acc, false, false);
      }
      const float bias = (j == 0) ? bl[m0] : 0.0f;      // bias hits scalar component only
      // stage to LDS (reuse XR region) as [r][m][j] so the final store is coalesced
      float* os = &lds[XR_OFF + (ksel * 8) * 512 + m0 * 8 + j];
      #pragma unroll
      for (int v = 0; v < 8; ++v) os[v * 512] = (acc[v] + bias) * inv_sqrt2;
    }
  }
  __syncthreads();

  // ---------- Phase F: coalesced tile store ----------
  {
    float4* og = (float4*)(out + (size_t)blockIdx.x * (TB * NOUT * DCOMP));
    #pragma unroll
    for (int it = 0; it < 8; ++it) {
      const int t = tid + it * NTHREADS;
      og[t] = *(const float4*)&lds[XR_OFF + t * 4];
    }
  }
}

extern "C" void kernel_launch(void* const* d_in, const int* in_sizes, int n_in,
                              void* d_out, int out_size, void* d_ws, size_t ws_size,
                              hipStream_t stream) {
  (void)in_sizes; (void)n_in; (void)out_size; (void)d_ws; (void)ws_size;
  const float* x   = (const float*)d_in[0];  // (8192,64,8)
  const float* wgt = (const float*)d_in[1];  // (64,64,4)
  const float* wr  = (const float*)d_in[2];  // (64,64,2)
  const float* wl  = (const float*)d_in[3];  // (64,64,2)
  const float* bl  = (const float*)d_in[4];  // (64,)
  const float* an  = (const float*)d_in[5];  // (64,2)
  float* out = (float*)d_out;

  const size_t shmem = LDS_FLOATS * sizeof(float);  // 128 KB dynamic LDS
  hipFuncSetAttribute((const void*)fcgp_kernel,
                      hipFuncAttributeMaxDynamicSharedMemorySize, (int)shmem);
  fcgp_kernel<<<BATCH / TB, NTHREADS, shmem, stream>>>(x, wgt, wr, wl, bl, an, out);
}